// QuantizerFunction_22892175687680
// MI455X (gfx1250) — compile-verified
//
#include <hip/hip_runtime.h>

#define LTOK   16384
#define DMODEL 512
#define HIDN   32
#define NLEVN  10

typedef __attribute__((ext_vector_type(2))) float v2f;
typedef __attribute__((ext_vector_type(8))) float v8f;

__device__ __forceinline__ v8f wmma4(v2f a, v2f b, v8f c) {
  // D = A(16x4,f32) * B(4x16,f32) + C(16x16,f32)
  return __builtin_amdgcn_wmma_f32_16x16x4_f32(false, a, false, b, (short)0, c, false, false);
}

__constant__ float c_FACT[NLEVN] = {1.f,1.f,1.f,2.f,2.f,2.f,4.f,4.f,4.f,32.f};
__constant__ float c_CBSZ[NLEVN] = {64.f,256.f,1024.f,64.f,256.f,1024.f,64.f,256.f,1024.f,9999.f};

// accum[0] = S2 = sum(s^2); accum[1] = sum of best scores; accum[2] = penalty sum
__global__ void __launch_bounds__(32) init_accum(float* accum) {
  if (threadIdx.x < 3) accum[threadIdx.x] = 0.f;
}

// s = state @ W_proj^T + b_proj  (L x 512 @ 512 x 32), WMMA f32 16x16x4, K-chained.
__global__ void __launch_bounds__(256) proj_kernel(
    const float* __restrict__ state, const float* __restrict__ Wp,
    const float* __restrict__ bp, float* __restrict__ s, float* __restrict__ accum) {
  const int wave = threadIdx.x >> 5, lane = threadIdx.x & 31;
  const int half = lane >> 4, lm = lane & 15, koff = half * 2;
  const int m0 = (blockIdx.x * 8 + wave) * 16;

  v8f c0 = {0.f,0.f,0.f,0.f,0.f,0.f,0.f,0.f};
  v8f c1 = {0.f,0.f,0.f,0.f,0.f,0.f,0.f,0.f};
  const float* arow  = state + (size_t)(m0 + lm) * DMODEL;
  const float* brow0 = Wp + (size_t)lm * DMODEL;          // W_proj[n, k] row-major -> B[k,n]=Wp[n*D+k]
  const float* brow1 = Wp + (size_t)(lm + 16) * DMODEL;
  for (int k0 = 0; k0 < DMODEL; k0 += 4) {
    v2f a  = *(const v2f*)(arow  + k0 + koff);
    v2f b0 = *(const v2f*)(brow0 + k0 + koff);
    v2f b1 = *(const v2f*)(brow1 + k0 + koff);
    c0 = wmma4(a, b0, c0);
    c1 = wmma4(a, b1, c1);
  }
  float bn0 = bp[lm], bn1 = bp[lm + 16];
  float ssum = 0.f;
  #pragma unroll
  for (int r = 0; r < 8; ++r) {
    int m = m0 + half * 8 + r;
    float v0 = c0[r] + bn0;
    float v1 = c1[r] + bn1;
    s[(size_t)m * HIDN + lm]      = v0;
    s[(size_t)m * HIDN + lm + 16] = v1;
    ssum += v0 * v0 + v1 * v1;
  }
  #pragma unroll
  for (int off = 16; off > 0; off >>= 1) ssum += __shfl_xor(ssum, off, 32);
  if (lane == 0) atomicAdd(&accum[0], ssum);
}

// Routing: tiny MHA -> per-head softmax over 10 levels -> mean -> +gumbel -> argmax.
// Writes one-hot att, level[], identity rows of mixed (level 9), penalty partials.
__global__ void __launch_bounds__(256) route_kernel(
    const float* __restrict__ s, const float* __restrict__ qkeys,
    const float* __restrict__ Wq, const float* __restrict__ bq,
    const float* __restrict__ Wk, const float* __restrict__ bk,
    const float* __restrict__ gum, int* __restrict__ level,
    float* __restrict__ att_out, float* __restrict__ mixed, float* __restrict__ accum) {
  __shared__ float lds_wq[HIDN * HIDN];
  __shared__ float lds_bq[HIDN];
  __shared__ float lds_kp[NLEVN * HIDN];
  const int tid = threadIdx.x;
  for (int i = tid; i < HIDN * HIDN; i += 256) lds_wq[i] = Wq[i];
  if (tid < HIDN) lds_bq[tid] = bq[tid];
  for (int i = tid; i < NLEVN * HIDN; i += 256) {
    int lev = i >> 5, j = i & 31;
    float v = bk[j];
    #pragma unroll
    for (int k = 0; k < HIDN; ++k) v += qkeys[lev * HIDN + k] * Wk[j * HIDN + k];
    lds_kp[i] = v;
  }
  __syncthreads();

  const int l = blockIdx.x * 256 + tid;
  float q[HIDN];
  #pragma unroll
  for (int k = 0; k < HIDN; ++k) q[k] = s[(size_t)l * HIDN + k];
  float qp[HIDN];
  #pragma unroll
  for (int j = 0; j < HIDN; ++j) {
    float v = lds_bq[j];
    #pragma unroll
    for (int k = 0; k < HIDN; ++k) v += q[k] * lds_wq[j * HIDN + k];
    qp[j] = v * 0.35355339059327373f;  // 1/sqrt(HEAD_DIM=8)
  }
  float att[NLEVN];
  #pragma unroll
  for (int lev = 0; lev < NLEVN; ++lev) att[lev] = 0.f;
  #pragma unroll
  for (int h = 0; h < 4; ++h) {
    float sc[NLEVN]; float mx = -3.4e38f;
    #pragma unroll
    for (int lev = 0; lev < NLEVN; ++lev) {
      float v = 0.f;
      #pragma unroll
      for (int d = 0; d < 8; ++d) v += qp[h * 8 + d] * lds_kp[lev * HIDN + h * 8 + d];
      sc[lev] = v; mx = fmaxf(mx, v);
    }
    float se = 0.f;
    #pragma unroll
    for (int lev = 0; lev < NLEVN; ++lev) { float e = expf(sc[lev] - mx); sc[lev] = e; se += e; }
    float inv = 1.f / se;
    #pragma unroll
    for (int lev = 0; lev < NLEVN; ++lev) att[lev] += sc[lev] * inv;
  }
  int best = 0; float bv = -3.4e38f;
  #pragma unroll
  for (int lev = 0; lev < NLEVN; ++lev) {
    float y = att[lev] * 0.25f + gum[(size_t)l * NLEVN + lev];  // softmax is monotone, TAU=1
    if (y > bv) { bv = y; best = lev; }
  }
  level[l] = best;
  #pragma unroll
  for (int lev = 0; lev < NLEVN; ++lev)
    att_out[(size_t)l * NLEVN + lev] = (lev == best) ? 1.f : 0.f;
  if (best == NLEVN - 1) {
    #pragma unroll
    for (int k = 0; k < HIDN; ++k) mixed[(size_t)l * HIDN + k] = q[k];
  }
  float p = c_FACT[best] + logf(c_CBSZ[best]);
  #pragma unroll
  for (int off = 16; off > 0; off >>= 1) p += __shfl_xor(p, off, 32);
  if ((tid & 31) == 0) atomicAdd(&accum[2], p);
}

// NN search for one codebook. d = 32/G, n embeds staged in LDS (k-major) + half-norms.
// WMMA computes X(16 x d) @ E(d x 16) score tiles; per-lane running argmax, then
// cross-lane reduction within each 16-lane half. BestSum accumulated; selected
// rows write the quantized vector into `mixed`.
template <int G>
__global__ void __launch_bounds__(256) quant_kernel(
    const float* __restrict__ s, const float* __restrict__ embed, int n, int cbidx,
    const int* __restrict__ level, float* __restrict__ mixed, float* __restrict__ accum) {
  constexpr int D_ = HIDN / G;
  constexpr int KS = D_ / 4;
  extern __shared__ float lds[];
  const int tid = threadIdx.x;
  for (int i = tid; i < D_ * n; i += 256) lds[i] = embed[i];
  for (int j = tid; j < n; j += 256) {
    float t = 0.f;
    #pragma unroll
    for (int k = 0; k < D_; ++k) { float e = embed[k * n + j]; t += e * e; }
    lds[D_ * n + j] = 0.5f * t;
  }
  __syncthreads();

  const int wave = tid >> 5, lane = tid & 31;
  const int half = lane >> 4, lm = lane & 15, koff = half * 2;
  const int v0 = (blockIdx.x * 8 + wave) * 16;

  const float* xbase = s + (size_t)v0 * D_;  // s reshaped (L*G, D_) is contiguous
  v2f afr[KS];
  #pragma unroll
  for (int kk = 0; kk < KS; ++kk)
    afr[kk] = *(const v2f*)(xbase + lm * D_ + kk * 4 + koff);

  float bestv[8]; int besti[8];
  #pragma unroll
  for (int r = 0; r < 8; ++r) { bestv[r] = -3.4e38f; besti[r] = 0; }

  for (int n0 = 0; n0 < n; n0 += 16) {
    v8f c = {0.f,0.f,0.f,0.f,0.f,0.f,0.f,0.f};
    #pragma unroll
    for (int kk = 0; kk < KS; ++kk) {
      v2f b;
      b.x = lds[(kk * 4 + koff) * n + n0 + lm];
      b.y = lds[(kk * 4 + koff + 1) * n + n0 + lm];
      c = wmma4(afr[kk], b, c);
    }
    float hn = lds[D_ * n + n0 + lm];
    int col = n0 + lm;
    #pragma unroll
    for (int r = 0; r < 8; ++r) {
      float sc = c[r] - hn;                 // x.e - |e|^2/2 ; argmax == argmin dist
      if (sc > bestv[r]) { bestv[r] = sc; besti[r] = col; }
    }
  }
  // reduce across the 16 lanes sharing the same 8 rows
  #pragma unroll
  for (int mask = 1; mask < 16; mask <<= 1) {
    #pragma unroll
    for (int r = 0; r < 8; ++r) {
      float ov = __shfl_xor(bestv[r], mask, 32);
      int   oi = __shfl_xor(besti[r], mask, 32);
      if (ov > bestv[r] || (ov == bestv[r] && oi < besti[r])) { bestv[r] = ov; besti[r] = oi; }
    }
  }
  float bsum = 0.f;
  #pragma unroll
  for (int r = 0; r < 8; ++r) bsum += bestv[r];
  float other = __shfl_xor(bsum, 16, 32);
  if (lane == 0) atomicAdd(&accum[1], bsum + other);

  // lanes 0..7 write rows 0..7; lanes 16..23 write rows 8..15 (compile-time reg indices)
  #pragma unroll
  for (int r = 0; r < 8; ++r) {
    if ((lane & 15) == r) {
      int v = v0 + half * 8 + r;
      int l = v / G;
      int grp = v - l * G;
      if (level[l] == cbidx) {
        int bi = besti[r];
        #pragma unroll
        for (int k = 0; k < D_; ++k)
          mixed[(size_t)l * HIDN + grp * D_ + k] = lds[k * n + bi];
      }
    }
  }
}

// out = mixed @ W_back^T + b_back  (L x 32 @ 32 x 512)
__global__ void __launch_bounds__(256) back_kernel(
    const float* __restrict__ mixed, const float* __restrict__ Wb,
    const float* __restrict__ bb, float* __restrict__ out) {
  const int wave = threadIdx.x >> 5, lane = threadIdx.x & 31;
  const int half = lane >> 4, lm = lane & 15, koff = half * 2;
  const int m0 = (blockIdx.x * 8 + wave) * 16;

  v2f afr[8];
  const float* arow = mixed + (size_t)(m0 + lm) * HIDN;
  #pragma unroll
  for (int kk = 0; kk < 8; ++kk) afr[kk] = *(const v2f*)(arow + kk * 4 + koff);

  for (int n0 = 0; n0 < DMODEL; n0 += 16) {
    v8f c = {0.f,0.f,0.f,0.f,0.f,0.f,0.f,0.f};
    const float* brow = Wb + (size_t)(n0 + lm) * HIDN;    // W_back[n, k]
    #pragma unroll
    for (int kk = 0; kk < 8; ++kk) {
      v2f b = *(const v2f*)(brow + kk * 4 + koff);
      c = wmma4(afr[kk], b, c);
    }
    float bn = bb[n0 + lm];
    #pragma unroll
    for (int r = 0; r < 8; ++r)
      out[(size_t)(m0 + half * 8 + r) * DMODEL + n0 + lm] = c[r] + bn;
  }
}

__global__ void __launch_bounds__(32) fin_kernel(const float* __restrict__ accum,
                                                 float* __restrict__ loss_out) {
  if (threadIdx.x == 0) {
    float S2 = accum[0], Best = accum[1], Pen = accum[2];
    float cb_loss = (9.f * S2 - 2.f * Best) / (10.f * (float)LTOK * 32.f);
    float penalty = Pen / (float)LTOK;
    loss_out[0] = cb_loss + 0.01f * penalty;
  }
}

extern "C" void kernel_launch(void* const* d_in, const int* in_sizes, int n_in,
                              void* d_out, int out_size, void* d_ws, size_t ws_size,
                              hipStream_t stream) {
  const float* state = (const float*)d_in[0];
  const float* Wp    = (const float*)d_in[1];
  const float* bp    = (const float*)d_in[2];
  const float* Wb    = (const float*)d_in[3];
  const float* bb    = (const float*)d_in[4];
  const float* qk    = (const float*)d_in[5];
  const float* Wq    = (const float*)d_in[6];
  const float* bq    = (const float*)d_in[7];
  const float* Wk    = (const float*)d_in[8];
  const float* bk    = (const float*)d_in[9];
  const float* gum   = (const float*)d_in[10];
  const float* cbs[9];
  for (int i = 0; i < 9; ++i) cbs[i] = (const float*)d_in[11 + i];

  float* out      = (float*)d_out;
  float* loss_out = out + (size_t)32 * 512 * 512;   // 8388608
  float* att_out  = loss_out + 1;

  float* s     = (float*)d_ws;
  float* mixed = s + (size_t)LTOK * HIDN;
  int*   level = (int*)(mixed + (size_t)LTOK * HIDN);
  float* accum = (float*)(level + LTOK);

  init_accum<<<1, 32, 0, stream>>>(accum);
  proj_kernel<<<128, 256, 0, stream>>>(state, Wp, bp, s, accum);
  route_kernel<<<LTOK / 256, 256, 0, stream>>>(s, qk, Wq, bq, Wk, bk, gum,
                                               level, att_out, mixed, accum);

  const int CBN[9] = {64, 256, 1024, 64, 256, 1024, 64, 256, 1024};
  for (int i = 0; i < 3; ++i) {
    size_t sh = (size_t)(32 * CBN[i] + CBN[i]) * sizeof(float);
    quant_kernel<1><<<128, 256, sh, stream>>>(s, cbs[i], CBN[i], i, level, mixed, accum);
  }
  for (int i = 3; i < 6; ++i) {
    size_t sh = (size_t)(16 * CBN[i] + CBN[i]) * sizeof(float);
    quant_kernel<2><<<256, 256, sh, stream>>>(s, cbs[i], CBN[i], i, level, mixed, accum);
  }
  for (int i = 6; i < 9; ++i) {
    size_t sh = (size_t)(8 * CBN[i] + CBN[i]) * sizeof(float);
    quant_kernel<4><<<512, 256, sh, stream>>>(s, cbs[i], CBN[i], i, level, mixed, accum);
  }
  back_kernel<<<128, 256, 0, stream>>>(mixed, Wb, bb, out);
  fin_kernel<<<1, 32, 0, stream>>>(accum, loss_out);
}